// SelfAttnTransformerLayer_34522947125324
// MI455X (gfx1250) — compile-verified
//
#include <hip/hip_runtime.h>
#include <hip/hip_bf16.h>

// ---------------------------------------------------------------------------
// SelfAttnTransformerLayer forward for MI455X (gfx1250, wave32, WMMA).
// f32 end-to-end (matches reference precision; the 268MB f32 prob output is
// the HBM roofline term at 23.3 TB/s, so lower-precision matrix ops can't
// win). Attention is fused (scores+softmax+probs-store+probs*V) so the 268MB
// prob tensor crosses HBM exactly once.
// ---------------------------------------------------------------------------

typedef __attribute__((ext_vector_type(2))) float v2f;
typedef __attribute__((ext_vector_type(8))) float v8f;

static constexpr int BATCH = 8;
static constexpr int AGENTS = 16;
static constexpr int TIME = 64;
static constexpr int H = 256;
static constexpr int NH = 8;
static constexpr int KD = 32;                  // key dim per head
static constexpr int FF = 128;
static constexpr int S = AGENTS * TIME;        // 1024 tokens per batch
static constexpr int MTOT = BATCH * S;         // 8192 rows total
static constexpr long long OUT_ELEMS = (long long)BATCH * S * H;   // 2,097,152
static constexpr long long TOK_ELEMS = (long long)MTOT * H;        // 2,097,152

__device__ __forceinline__ v8f wmma_f32_16x16x4(v2f a, v2f b, v8f c) {
    return __builtin_amdgcn_wmma_f32_16x16x4_f32(
        false, a, false, b, (short)0, c, false, false);
}

// ---------------------------------------------------------------------------
// Plain f32 WMMA GEMM: one wave per 16x16 tile of
//   C = A[M,K] x B[K,N] + bias[N]   (+ optional ReLU)
// Used for QKV / out-proj / FFN (weights + activations are L2-resident).
// ---------------------------------------------------------------------------
template <bool RELU>
__global__ __launch_bounds__(32) void wmma_gemm_f32(
    const float* __restrict__ A, const float* __restrict__ B,
    float* __restrict__ C, const float* __restrict__ bias,
    int K, int lda, int ldb, int ldc)
{
    const int tn   = blockIdx.x * 16;
    const int tm   = blockIdx.y * 16;
    const int lane = threadIdx.x;       // wave32
    const int hl   = lane >> 4;         // 0: K pair {k,k+1}, 1: {k+2,k+3}
    const int l15  = lane & 15;

    const int arow = tm + l15;
    const int bcol = tn + l15;

    v8f acc = {};
    for (int k0 = 0; k0 < K; k0 += 4) {
        const int ka = k0 + hl * 2;
        v2f afrag = *(const v2f*)(A + (long long)arow * lda + ka);
        v2f bfrag;
        bfrag.x = B[(long long)ka * ldb + bcol];
        bfrag.y = B[(long long)(ka + 1) * ldb + bcol];
        acc = wmma_f32_16x16x4(afrag, bfrag, acc);
    }

    const float bv = bias[bcol];
    const int row0 = tm + hl * 8;       // D layout: VGPR i -> rows tm+i / tm+8+i
#pragma unroll
    for (int i = 0; i < 8; ++i) {
        float v = acc[i] + bv;
        if (RELU) v = fmaxf(v, 0.0f);
        C[(long long)(row0 + i) * ldc + bcol] = v;
    }
}

// ---------------------------------------------------------------------------
// Fused attention per (batch, head, 16-row block): 8 waves / 256 threads.
//   wave w owns score columns [w*128, w*128+128)
//   Phase 1: scores (WMMA, K=32), Phase 2: softmax (shfl + LDS),
//   Phase 3: write probs to d_out + swizzled LDS stash,
//   Phase 4: ctx partial = P_strip x V_strip (WMMA, A from own LDS stash),
//            cross-wave reduction through LDS.
// LDS bank-conflict avoidance: column swizzle c ^ ((2*row)&31) keeps the
// 16-lane A-fragment b64 reads on 16 distinct bank pairs.
// ---------------------------------------------------------------------------
__global__ __launch_bounds__(256) void fused_attention(
    const float* __restrict__ Q, const float* __restrict__ Km,
    const float* __restrict__ V, float* __restrict__ probs,
    float* __restrict__ ctx)
{
    __shared__ float lds[16384];        // 64KB -> 4+ workgroups per 320KB WGP

    const int m0   = blockIdx.x * 16;   // query-row block
    const int n    = blockIdx.y;        // head
    const int b    = blockIdx.z;        // batch
    const int wave = threadIdx.x >> 5;
    const int lane = threadIdx.x & 31;
    const int hl   = lane >> 4;
    const int l15  = lane & 15;

    const float* Qh = Q  + (long long)b * S * H + n * KD;
    const float* Kh = Km + (long long)b * S * H + n * KD;
    const float* Vh = V  + (long long)b * S * H + n * KD;
    float* prow = probs + (((long long)b * NH + n) * S + m0) * S;

    // ---- Phase 1: score strip, cols [wave*128, wave*128+128), K = 32 ----
    v2f qf[8];
#pragma unroll
    for (int k = 0; k < 8; ++k)
        qf[k] = *(const v2f*)(Qh + (long long)(m0 + l15) * H + k * 4 + hl * 2);

    v8f acc[8];
#pragma unroll
    for (int t = 0; t < 8; ++t) acc[t] = (v8f){};

#pragma unroll
    for (int t = 0; t < 8; ++t) {
        const float* kp = Kh + (long long)(wave * 128 + t * 16 + l15) * H;
#pragma unroll
        for (int k = 0; k < 8; ++k) {
            v2f bf = *(const v2f*)(kp + k * 4 + hl * 2);
            acc[t] = wmma_f32_16x16x4(qf[k], bf, acc[t]);
        }
    }

    const float scale = 0.17677669529663687f;   // 1/sqrt(32)
#pragma unroll
    for (int t = 0; t < 8; ++t)
#pragma unroll
        for (int i = 0; i < 8; ++i) acc[t][i] *= scale;

    // ---- Phase 2: softmax over 1024 columns ----
    // VGPR idx i holds row (i + hl*8); xor 1/2/4/8 stays within a 16-lane half.
    float lmax[8];
#pragma unroll
    for (int i = 0; i < 8; ++i) {
        float m = acc[0][i];
#pragma unroll
        for (int t = 1; t < 8; ++t) m = fmaxf(m, acc[t][i]);
        lmax[i] = m;
    }
#pragma unroll
    for (int off = 1; off < 16; off <<= 1)
#pragma unroll
        for (int i = 0; i < 8; ++i)
            lmax[i] = fmaxf(lmax[i], __shfl_xor(lmax[i], off, 32));

    if (l15 == 0) {
#pragma unroll
        for (int i = 0; i < 8; ++i) lds[wave * 16 + hl * 8 + i] = lmax[i];
    }
    __syncthreads();
    float rmax[8];
#pragma unroll
    for (int i = 0; i < 8; ++i) {
        float m = lds[hl * 8 + i];
#pragma unroll
        for (int w = 1; w < 8; ++w) m = fmaxf(m, lds[w * 16 + hl * 8 + i]);
        rmax[i] = m;
    }
    __syncthreads();

    float lsum[8];
#pragma unroll
    for (int i = 0; i < 8; ++i) lsum[i] = 0.0f;
#pragma unroll
    for (int t = 0; t < 8; ++t)
#pragma unroll
        for (int i = 0; i < 8; ++i) {
            float e = __expf(acc[t][i] - rmax[i]);
            acc[t][i] = e;
            lsum[i] += e;
        }
#pragma unroll
    for (int off = 1; off < 16; off <<= 1)
#pragma unroll
        for (int i = 0; i < 8; ++i) lsum[i] += __shfl_xor(lsum[i], off, 32);

    if (l15 == 0) {
#pragma unroll
        for (int i = 0; i < 8; ++i) lds[wave * 16 + hl * 8 + i] = lsum[i];
    }
    __syncthreads();
    float rinv[8];
#pragma unroll
    for (int i = 0; i < 8; ++i) {
        float s = lds[hl * 8 + i];
#pragma unroll
        for (int w = 1; w < 8; ++w) s += lds[w * 16 + hl * 8 + i];
        rinv[i] = 1.0f / s;
    }
    __syncthreads();    // all waves done with the 128-float scratch region

    // ---- Phase 3: normalize; write probs to global + swizzled LDS stash ----
#pragma unroll
    for (int t = 0; t < 8; ++t) {
#pragma unroll
        for (int i = 0; i < 8; ++i) {
            const float p  = acc[t][i] * rinv[i];
            const int row  = i + hl * 8;
            const int c    = t * 16 + l15;                    // local col 0..127
            prow[(long long)row * S + wave * 128 + c] = p;
            lds[wave * 2048 + row * 128 + (c ^ ((row << 1) & 31))] = p;
        }
    }
    // No barrier needed: phase 4 reads only this wave's own stash
    // (the wave's K-range equals the prob columns it just produced).

    // ---- Phase 4: ctx partial = P[16 x 128] x V[128 x 32] ----
    v8f cacc[2];
    cacc[0] = (v8f){};
    cacc[1] = (v8f){};
    const float* vbase = Vh + (long long)(wave * 128) * H;
    const float* my = &lds[wave * 2048];
    const int swz = (l15 << 1) & 31;                 // A-frag row = l15
    for (int k = 0; k < 128; k += 4) {
        const int kk = k + hl * 2;
        v2f af = *(const v2f*)(my + l15 * 128 + (kk ^ swz));
        v2f bf0, bf1;
        bf0.x = vbase[(long long)kk * H + l15];
        bf0.y = vbase[(long long)(kk + 1) * H + l15];
        bf1.x = vbase[(long long)kk * H + 16 + l15];
        bf1.y = vbase[(long long)(kk + 1) * H + 16 + l15];
        cacc[0] = wmma_f32_16x16x4(af, bf0, cacc[0]);
        cacc[1] = wmma_f32_16x16x4(af, bf1, cacc[1]);
    }
    __syncthreads();    // everyone done reading stash; reuse lds for reduction

    // Cross-wave reduction of the 2 ctx tiles (16x32 total).
#pragma unroll
    for (int t = 0; t < 2; ++t)
#pragma unroll
        for (int i = 0; i < 8; ++i)
            lds[wave * 512 + t * 256 + i * 32 + lane] = cacc[t][i];
    __syncthreads();

#pragma unroll
    for (int p = 0; p < 2; ++p) {
        const int pos = p * 256 + threadIdx.x;       // t*256 + i*32 + ln
        const int t  = pos >> 8;
        const int i  = (pos >> 5) & 7;
        const int ln = pos & 31;
        float s = 0.0f;
#pragma unroll
        for (int w = 0; w < 8; ++w) s += lds[w * 512 + pos];
        const int row  = m0 + i + (ln >> 4) * 8;
        const int colh = t * 16 + (ln & 15);
        ctx[((long long)b * S + row) * H + n * KD + colh] = s;
    }
}

// ---------------------------------------------------------------------------
// out[tok] = LayerNorm(a[tok] + r[tok]) * g + beta   (H = 256 = blockDim)
// ---------------------------------------------------------------------------
__global__ __launch_bounds__(256) void ln_residual(
    const float* __restrict__ a, const float* __restrict__ r,
    const float* __restrict__ g, const float* __restrict__ beta,
    float* __restrict__ out)
{
    __shared__ float red[256];
    const long long base = (long long)blockIdx.x * 256;
    const int tid = threadIdx.x;

    const float v = a[base + tid] + r[base + tid];
    red[tid] = v;
    __syncthreads();
#pragma unroll
    for (int s = 128; s > 0; s >>= 1) {
        if (tid < s) red[tid] += red[tid + s];
        __syncthreads();
    }
    const float mu = red[0] * (1.0f / 256.0f);
    __syncthreads();

    const float d = v - mu;
    red[tid] = d * d;
    __syncthreads();
#pragma unroll
    for (int s = 128; s > 0; s >>= 1) {
        if (tid < s) red[tid] += red[tid + s];
        __syncthreads();
    }
    const float var = red[0] * (1.0f / 256.0f);
    const float rs = rsqrtf(var + 1e-6f);
    out[base + tid] = d * rs * g[tid] + beta[tid];
}

// ---------------------------------------------------------------------------
extern "C" void kernel_launch(void* const* d_in, const int* in_sizes, int n_in,
                              void* d_out, int out_size, void* d_ws, size_t ws_size,
                              hipStream_t stream)
{
    (void)in_sizes; (void)n_in; (void)out_size; (void)ws_size;

    const float* x   = (const float*)d_in[0];   // [B,S,H]
    const float* wq  = (const float*)d_in[1];   // [H, NH*KD]
    const float* bq  = (const float*)d_in[2];
    const float* wk  = (const float*)d_in[3];
    const float* bk  = (const float*)d_in[4];
    const float* wv  = (const float*)d_in[5];
    const float* bv  = (const float*)d_in[6];
    const float* wo  = (const float*)d_in[7];   // [NH*KD, H]
    const float* bo  = (const float*)d_in[8];
    const float* g1  = (const float*)d_in[9];
    const float* be1 = (const float*)d_in[10];
    const float* g2  = (const float*)d_in[11];
    const float* be2 = (const float*)d_in[12];
    const float* w1  = (const float*)d_in[13];  // [H, FF]
    const float* b1  = (const float*)d_in[14];
    const float* w2  = (const float*)d_in[15];  // [FF, H]
    const float* b2  = (const float*)d_in[16];

    float* out  = (float*)d_out;                // [B,A,T,H]
    float* attn = out + OUT_ELEMS;              // [B,NH,S,S]

    // Four 8MB workspace regions (32MB total):
    //   R0: Q   -> ff1
    //   R1: K   -> proj -> z
    //   R2: V   -> y
    //   R3: ctx
    float* R0 = (float*)d_ws;
    float* R1 = R0 + TOK_ELEMS;
    float* R2 = R1 + TOK_ELEMS;
    float* R3 = R2 + TOK_ELEMS;

    const dim3 blk(32);

    // 1) QKV projections: [8192,256] x [256,256] + bias
    {
        dim3 grid(H / 16, MTOT / 16, 1);
        wmma_gemm_f32<false><<<grid, blk, 0, stream>>>(x, wq, R0, bq, H, H, H, H);
        wmma_gemm_f32<false><<<grid, blk, 0, stream>>>(x, wk, R1, bk, H, H, H, H);
        wmma_gemm_f32<false><<<grid, blk, 0, stream>>>(x, wv, R2, bv, H, H, H, H);
    }

    // 2) fused attention: probs -> d_out (single HBM pass), ctx -> R3
    {
        dim3 grid(S / 16, NH, BATCH);
        fused_attention<<<grid, 256, 0, stream>>>(R0, R1, R2, attn, R3);
    }

    // 3) out-proj: ctx x Wo + bo -> R1 (K dead)
    {
        dim3 grid(H / 16, MTOT / 16, 1);
        wmma_gemm_f32<false><<<grid, blk, 0, stream>>>(R3, wo, R1, bo, H, H, H, H);
    }

    // 4) y = LN(proj + x) -> R2 (V dead)
    ln_residual<<<MTOT, 256, 0, stream>>>(R1, x, g1, be1, R2);

    // 5) ff = relu(y x W1 + b1) -> R0 (Q dead)      [8192,256]x[256,128]
    {
        dim3 grid(FF / 16, MTOT / 16, 1);
        wmma_gemm_f32<true><<<grid, blk, 0, stream>>>(R2, w1, R0, b1, H, H, FF, FF);
    }

    // 6) z = ff x W2 + b2 -> R1 (proj dead)         [8192,128]x[128,256]
    {
        dim3 grid(H / 16, MTOT / 16, 1);
        wmma_gemm_f32<false><<<grid, blk, 0, stream>>>(R0, w2, R1, b2, FF, FF, H, H);
    }

    // 7) out = LN(z + y) -> d_out
    ln_residual<<<MTOT, 256, 0, stream>>>(R1, R2, g2, be2, out);
}